// ModuleCorrelation_17282948399396
// MI455X (gfx1250) — compile-verified
//
#include <hip/hip_runtime.h>
#include <hip/hip_bf16.h>

typedef __attribute__((ext_vector_type(2))) float v2f;
typedef __attribute__((ext_vector_type(8))) float v8f;

#define BATCH 8
#define CH    128
#define HH    128
#define WW    256
#define PAD   4
#define NDISP 9            // 2*PAD+1
#define HW    (HH * WW)

// One wave32 handles a 16-pixel horizontal strip (b, y, x0..x0+15) and
// produces all 81 correlation outputs for those pixels.
//   A (16x128)  = ref[b, :, y, x0:x0+16]
//   per dy: B (128x32) = qry[b, :, y+dy-4, x0-4 : x0+28]  (zero outside)
//   D (16x32) band n = m + dx, dx in [0,8] -> out[b, dy*9+dx, y, x0+m]
__global__ __launch_bounds__(128) void corr_wmma_f32_kernel(
    const float* __restrict__ ref,
    const float* __restrict__ qry,
    float* __restrict__ out)
{
    const int lane = threadIdx.x;                 // 0..31
    const int wid  = blockIdx.x * 4 + threadIdx.y;
    const int xt   = wid & 15;                    // W/16 = 16 tiles
    const int y    = (wid >> 4) & (HH - 1);
    const int b    = wid >> 11;                   // / (16*128)
    const int x0   = xt << 4;

    const int m     = lane & 15;                  // M index (pixel in strip)
    const int khalf = (lane >> 4) << 1;           // K sub-offset: 0 or 2

    const float* refb = ref + (size_t)b * CH * HW;
    const float* qb   = qry + (size_t)b * CH * HW;

    // ---- preload A fragments: 32 K-steps x (2 VGPR) = 16 px x 128 ch ----
    v2f a[32];
    const int aoff = y * WW + x0 + m;
#pragma unroll
    for (int ks = 0; ks < 32; ++ks) {
        const int c0 = 4 * ks + khalf;
        a[ks].x = refb[c0 * HW + aoff];           // K = 4ks+khalf
        a[ks].y = refb[(c0 + 1) * HW + aoff];     // K = 4ks+khalf+1
    }

    // dy-invariant x addressing for the two B tiles (N=0..15 and N=16..31).
    // Addresses are CLAMPED so every load is valid -> loads can issue
    // unconditionally; out-of-range lanes are zeroed with v_cndmask after.
    const int  xq0  = x0 - PAD + m;               // tile0 query column
    const int  xq1  = x0 + 12 + m;                // tile1 query column
    const bool xin0 = (xq0 >= 0);                 // right side always < W
    const bool xin1 = (xq1 < WW);
    const int  xc0  = xin0 ? xq0 : 0;
    const int  xc1  = xin1 ? xq1 : (WW - 1);

    const float scale = 1.0f / (float)CH;
    const int   mBase = (lane < 16) ? 0 : 8;      // C/D row half per lane
    const int   n0    = lane & 15;                // C/D column per lane

#pragma unroll 1
    for (int dy = 0; dy < NDISP; ++dy) {
        const int  yq    = y + dy - PAD;
        const bool yin   = (yq >= 0) && (yq < HH);
        const float* pdy = qb + (yin ? yq : 0) * WW;   // uniform per dy

        const bool mask0 = yin && xin0;           // per-lane, computed once
        const bool mask1 = yin && xin1;

        v8f acc0 = {0.f, 0.f, 0.f, 0.f, 0.f, 0.f, 0.f, 0.f};
        v8f acc1 = {0.f, 0.f, 0.f, 0.f, 0.f, 0.f, 0.f, 0.f};

#pragma unroll
        for (int ks = 0; ks < 32; ++ks) {
            const int    c0 = 4 * ks + khalf;
            const float* p0 = pdy + c0 * HW;
            const float* p1 = pdy + (c0 + 1) * HW;

            // unconditional loads (always-valid clamped addresses)
            const float t00 = p0[xc0];
            const float t01 = p1[xc0];
            const float t10 = p0[xc1];
            const float t11 = p1[xc1];

            v2f bt0, bt1;                          // zero edges via cndmask
            bt0.x = mask0 ? t00 : 0.0f;
            bt0.y = mask0 ? t01 : 0.0f;
            bt1.x = mask1 ? t10 : 0.0f;
            bt1.y = mask1 ? t11 : 0.0f;

            acc0 = __builtin_amdgcn_wmma_f32_16x16x4_f32(
                       false, a[ks], false, bt0, (short)0, acc0, false, false);
            acc1 = __builtin_amdgcn_wmma_f32_16x16x4_f32(
                       false, a[ks], false, bt1, (short)0, acc1, false, false);
        }

        // scatter the useful 16x9 band out of the 16x32 D tile
#pragma unroll
        for (int t = 0; t < 2; ++t) {
            const v8f acc = t ? acc1 : acc0;
            const int Ng  = n0 + t * 16;
#pragma unroll
            for (int r = 0; r < 8; ++r) {
                const int mm  = mBase + r;
                const int dxv = Ng - mm;
                if (dxv >= 0 && dxv < NDISP) {
                    const int d = dy * NDISP + dxv;
                    out[(((size_t)b * (NDISP * NDISP) + d) * HH + y) * WW + x0 + mm] =
                        acc[r] * scale;
                }
            }
        }
    }
}

extern "C" void kernel_launch(void* const* d_in, const int* in_sizes, int n_in,
                              void* d_out, int out_size, void* d_ws, size_t ws_size,
                              hipStream_t stream) {
    const float* ref = (const float*)d_in[0];   // reference_features
    const float* qry = (const float*)d_in[1];   // query_features
    float* out = (float*)d_out;                 // (B, 81, H, W) fp32

    // one wave per (b, y, 16-px x-tile): 8*128*16 = 16384 waves, 4 waves/block
    dim3 block(32, 4);
    dim3 grid((BATCH * HH * (WW / 16)) / 4);
    corr_wmma_f32_kernel<<<grid, block, 0, stream>>>(ref, qry, out);
}